// SageLP_69630009802898
// MI455X (gfx1250) — compile-verified
//
#include <hip/hip_runtime.h>

#define NNODES 50000
#define NEDGES 800000
#define FEAT   128          // IN_C == HID == HID1 == 128 (all scattered features)

typedef __attribute__((ext_vector_type(2))) float v2f;
typedef __attribute__((ext_vector_type(8))) float v8f;

// ---------------- utility: zero fill ----------------
__global__ void zero_kernel(float* __restrict__ p, long long n) {
    long long i = (long long)blockIdx.x * blockDim.x + threadIdx.x;
    long long stride = (long long)gridDim.x * blockDim.x;
    for (; i < n; i += stride) p[i] = 0.0f;
}

// ---------------- degree: atomic count per dst ----------------
__global__ void deg_count_kernel(const int* __restrict__ dst, float* __restrict__ deg) {
    int e = blockIdx.x * blockDim.x + threadIdx.x;
    if (e < NEDGES) atomicAdd(&deg[dst[e]], 1.0f);
}

__global__ void deg_inv_kernel(float* __restrict__ deg) {
    int i = blockIdx.x * blockDim.x + threadIdx.x;
    if (i < NNODES) deg[i] = 1.0f / fmaxf(deg[i], 1.0f);
}

// ---------------- edge scatter-add: agg[dst] += feat[src] ----------------
// FEAT=128 -> 32 float4 chunks per edge; one thread per (edge, chunk).
__global__ void scatter_add_kernel(const float* __restrict__ feat,
                                   const int* __restrict__ src,
                                   const int* __restrict__ dst,
                                   float* __restrict__ agg) {
    long long tid = (long long)blockIdx.x * blockDim.x + threadIdx.x;
    const long long total = (long long)NEDGES * (FEAT / 4);
    if (tid >= total) return;
    int e    = (int)(tid >> 5);        // FEAT/4 == 32
    int cblk = (int)(tid & 31);
    int s = src[e];
    int d = dst[e];
    const float4 v = ((const float4*)(feat + (long long)s * FEAT))[cblk];
    float* o = agg + (long long)d * FEAT + (cblk << 2);
    atomicAdd(o + 0, v.x);
    atomicAdd(o + 1, v.y);
    atomicAdd(o + 2, v.z);
    atomicAdd(o + 3, v.w);
}

// ---------------- fused SAGE layer GEMM via V_WMMA_F32_16X16X4_F32 -------
// out[n,:] = act( (agg[n,:]*dinv[n]) @ Wl + xin[n,:] @ Wr + bias )
// One wave (32 lanes) computes one 16x16 output tile. wave32, K step = 4.
//
// fp32 A fragment (16x4):  lane<16: v0=A[l,k+0] v1=A[l,k+1]
//                          lane>=16: v0=A[l-16,k+2] v1=A[l-16,k+3]
// fp32 B fragment (4x16):  mirrored (K split by lane half, N = lane&15)
// fp32 C fragment (16x16): VGPR j -> row j (lanes 0-15) / row j+8 (lanes 16-31)
__global__ void sage_gemm_wmma_kernel(const float* __restrict__ agg,
                                      const float* __restrict__ xin,
                                      const float* __restrict__ dinv,
                                      const float* __restrict__ Wl,
                                      const float* __restrict__ Wr,
                                      const float* __restrict__ bias,
                                      float* __restrict__ out,
                                      int Fin, int Fout, int do_relu) {
    const int lane = threadIdx.x;                 // 0..31 (wave32)
    const int wave = threadIdx.y;                 // 0..7
    const int ntn  = Fout >> 4;                   // N tiles
    const int tile = blockIdx.x * blockDim.y + wave;
    const int m_tile = tile / ntn;
    const int n_tile = tile - m_tile * ntn;
    if (m_tile >= (NNODES >> 4)) return;          // wave-uniform exit: EXEC stays all-1s

    const int half = lane >> 4;                   // 0: K+0/1, 1: K+2/3
    const int l15  = lane & 15;
    const int row  = (m_tile << 4) + l15;         // A row held by this lane
    const int col  = (n_tile << 4) + l15;         // B / output column held by this lane
    const float di = dinv[row];

    const float* __restrict__ arow = agg + (long long)row * Fin;
    const float* __restrict__ xrow = xin + (long long)row * Fin;

    v8f c = {0.f, 0.f, 0.f, 0.f, 0.f, 0.f, 0.f, 0.f};

    // mean part:  (agg*dinv) @ Wl
    for (int k = 0; k < Fin; k += 4) {
        const int ka = k + (half << 1);
        v2f a, b;
        a.x = arow[ka]     * di;
        a.y = arow[ka + 1] * di;
        b.x = Wl[(long long)ka       * Fout + col];
        b.y = Wl[(long long)(ka + 1) * Fout + col];
        c = __builtin_amdgcn_wmma_f32_16x16x4_f32(false, a, false, b,
                                                  (short)0, c, false, false);
    }
    // self part:  x @ Wr
    for (int k = 0; k < Fin; k += 4) {
        const int ka = k + (half << 1);
        v2f a, b;
        a.x = xrow[ka];
        a.y = xrow[ka + 1];
        b.x = Wr[(long long)ka       * Fout + col];
        b.y = Wr[(long long)(ka + 1) * Fout + col];
        c = __builtin_amdgcn_wmma_f32_16x16x4_f32(false, a, false, b,
                                                  (short)0, c, false, false);
    }

    const float bn = bias[col];
#pragma unroll
    for (int j = 0; j < 8; ++j) {
        const int m = (m_tile << 4) + j + (half << 3);   // rows j / j+8
        float v = c[j] + bn;
        if (do_relu) v = fmaxf(v, 0.0f);
        out[(long long)m * Fout + col] = v;
    }
}

// ---------------- host-side orchestration ----------------
static inline void run_layer(const float* agg, const float* xin, const float* dinv,
                             const float* Wl, const float* Wr, const float* b,
                             float* out, int Fout, int do_relu, hipStream_t stream) {
    const int tiles = (NNODES / 16) * (Fout / 16);
    dim3 blk(32, 8);
    dim3 grd((tiles + 7) / 8);
    sage_gemm_wmma_kernel<<<grd, blk, 0, stream>>>(agg, xin, dinv, Wl, Wr, b, out,
                                                   FEAT, Fout, do_relu);
}

extern "C" void kernel_launch(void* const* d_in, const int* in_sizes, int n_in,
                              void* d_out, int out_size, void* d_ws, size_t ws_size,
                              hipStream_t stream) {
    const float* x   = (const float*)d_in[0];
    const int*   ei  = (const int*)  d_in[1];   // [2, E]
    const float* Wl1 = (const float*)d_in[2];
    const float* Wr1 = (const float*)d_in[3];
    const float* b1  = (const float*)d_in[4];
    const float* Wl2 = (const float*)d_in[5];
    const float* Wr2 = (const float*)d_in[6];
    const float* b2  = (const float*)d_in[7];
    const float* Wl3 = (const float*)d_in[8];
    const float* Wr3 = (const float*)d_in[9];
    const float* b3  = (const float*)d_in[10];

    const int* src = ei;
    const int* dst = ei + NEDGES;

    // workspace layout (floats): deg[N] | agg[N*128] | h1[N*128] | h2[N*128]
    float* ws  = (float*)d_ws;
    float* deg = ws;
    float* agg = deg + NNODES;
    float* h1  = agg + (long long)NNODES * FEAT;
    float* h2  = h1  + (long long)NNODES * FEAT;

    const long long nfeat = (long long)NNODES * FEAT;
    const int Z = 256;

    // degrees -> deg_inv (in place)
    zero_kernel<<<1024, Z, 0, stream>>>(deg, NNODES);
    deg_count_kernel<<<(NEDGES + Z - 1) / Z, Z, 0, stream>>>(dst, deg);
    deg_inv_kernel<<<(NNODES + Z - 1) / Z, Z, 0, stream>>>(deg);

    const long long sc_threads = (long long)NEDGES * (FEAT / 4);
    const int sc_blocks = (int)((sc_threads + Z - 1) / Z);

    // layer 1: h1 = relu(mean(x) @ Wl1 + x @ Wr1 + b1)
    zero_kernel<<<4096, Z, 0, stream>>>(agg, nfeat);
    scatter_add_kernel<<<sc_blocks, Z, 0, stream>>>(x, src, dst, agg);
    run_layer(agg, x, deg, Wl1, Wr1, b1, h1, 128, 1, stream);

    // layer 2: h2 = relu(mean(h1) @ Wl2 + h1 @ Wr2 + b2)
    zero_kernel<<<4096, Z, 0, stream>>>(agg, nfeat);
    scatter_add_kernel<<<sc_blocks, Z, 0, stream>>>(h1, src, dst, agg);
    run_layer(agg, h1, deg, Wl2, Wr2, b2, h2, 128, 1, stream);

    // layer 3: out = mean(h2) @ Wl3 + h2 @ Wr3 + b3   (no relu)
    zero_kernel<<<4096, Z, 0, stream>>>(agg, nfeat);
    scatter_add_kernel<<<sc_blocks, Z, 0, stream>>>(h2, src, dst, agg);
    run_layer(agg, h2, deg, Wl3, Wr3, b3, (float*)d_out, 64, 0, stream);
}